// MaskGEVBeamformer_68607807586623
// MI455X (gfx1250) — compile-verified
//
#include <hip/hip_runtime.h>
#include <math.h>

// ---------------------------------------------------------------------------
// GEV mask beamformer for MI455X (gfx1250, wave32, WMMA).
// Phase 1: covariance via V_WMMA_F32_16X16X4_F32 (real/imag stacked 16xT),
//          LDS-staged in 200-frame chunks + global_prefetch of next chunk.
// Phase 2: per-(b,s,f) 8x8 complex Cholesky + Jacobi eigensolver (1 thread ea).
// Phase 3: streamed beamformer apply, b128 loads/stores (memory bound).
// ---------------------------------------------------------------------------

typedef float v2f __attribute__((ext_vector_type(2)));
typedef float v4f __attribute__((ext_vector_type(4)));
typedef float v8f __attribute__((ext_vector_type(8)));

constexpr int Bn = 8;      // batch
constexpr int Cn = 8;      // mics
constexpr int Fn = 257;    // freq bins
constexpr int Tn = 1000;   // frames
constexpr int Sn = 2;      // sources

constexpr int TCH = 200;   // T-chunk staged in LDS (divides 1000, mult of 4)
constexpr int SST = 204;   // LDS row stride (floats): 204*4B = 16B aligned,
                           // 204 mod 64 = 12 -> rows 0..15 hit distinct banks

// ========================== Phase 1: covariance ============================
// One wave per (b,f). Accumulators:
//   acc0 = s0 target, acc1 = s0 noise, acc2 = s1 target, acc3 = s1 noise
// Each is M = (S*diag(w)) * S^T, S = [Xr rows 0..7; Xi rows 8..15] (16 x T).
__global__ __launch_bounds__(32)
void cov_wmma_kernel(const float* __restrict__ Xr,
                     const float* __restrict__ Xi,
                     const float* __restrict__ Mk,
                     float* __restrict__ Rws)
{
    __shared__ float sm[16 * SST + 2 * TCH];   // ~14.7 KB
    float* sM0 = sm + 16 * SST;
    float* sM1 = sM0 + TCH;

    const int bf   = blockIdx.x;
    const int b    = bf / Fn;
    const int f    = bf % Fn;
    const int lane = threadIdx.x;

    v8f acc0 = {}, acc1 = {}, acc2 = {}, acc3 = {};

    const int row = lane & 15;            // stacked-matrix row for this lane
    const int tsh = (lane >> 4) << 1;     // K sub-offset: lanes 16..31 -> +2

    for (int ch = 0; ch < Tn / TCH; ++ch) {
        const int t0c = ch * TCH;

        // ---- stage X rows (real + imag) and both masks into LDS ----
        for (int c = 0; c < Cn; ++c) {
            const float4* gr = (const float4*)(Xr + ((size_t)(b * Cn + c) * Fn + f) * Tn + t0c);
            const float4* gi = (const float4*)(Xi + ((size_t)(b * Cn + c) * Fn + f) * Tn + t0c);
            for (int q = lane; q < TCH / 4; q += 32) {
                *(float4*)&sm[c * SST + 4 * q]       = gr[q];
                *(float4*)&sm[(c + 8) * SST + 4 * q] = gi[q];
            }
        }
        {
            const float4* g0 = (const float4*)(Mk + ((size_t)(b * Sn + 0) * Fn + f) * Tn + t0c);
            const float4* g1 = (const float4*)(Mk + ((size_t)(b * Sn + 1) * Fn + f) * Tn + t0c);
            for (int q = lane; q < TCH / 4; q += 32) {
                *(float4*)&sM0[4 * q] = g0[q];
                *(float4*)&sM1[4 * q] = g1[q];
            }
        }
        __syncthreads();

        // ---- prefetch next chunk into cache while we compute this one ----
        if (ch + 1 < Tn / TCH) {
            const int tnx = t0c + TCH;
            const int po  = lane * 8;     // 32 lanes x 32B spacing covers 800B row
            for (int c = 0; c < Cn; ++c) {
                __builtin_prefetch(Xr + ((size_t)(b * Cn + c) * Fn + f) * Tn + tnx + po, 0, 3);
                __builtin_prefetch(Xi + ((size_t)(b * Cn + c) * Fn + f) * Tn + tnx + po, 0, 3);
            }
            __builtin_prefetch(Mk + ((size_t)(b * Sn + 0) * Fn + f) * Tn + tnx + po, 0, 3);
            __builtin_prefetch(Mk + ((size_t)(b * Sn + 1) * Fn + f) * Tn + tnx + po, 0, 3);
        }

        // ---- WMMA accumulation over this chunk (K=4 per step) ----
        for (int k = 0; k < TCH / 4; ++k) {
            const int ta = 4 * k + tsh;
            const float2 base = *(const float2*)&sm[row * SST + ta];
            const float2 m0   = *(const float2*)&sM0[ta];
            const float2 m1   = *(const float2*)&sM1[ta];

            v2f bfr; bfr.x = base.x; bfr.y = base.y;   // shared B fragment
            v2f a;

            // source 0 target: w = m0^2
            a.x = base.x * (m0.x * m0.x);
            a.y = base.y * (m0.y * m0.y);
            acc0 = __builtin_amdgcn_wmma_f32_16x16x4_f32(false, a, false, bfr,
                                                         (short)0, acc0, false, false);
            // source 0 noise: w = (1-m0)^2
            {
                float u0 = 1.f - m0.x, u1 = 1.f - m0.y;
                a.x = base.x * (u0 * u0);
                a.y = base.y * (u1 * u1);
            }
            acc1 = __builtin_amdgcn_wmma_f32_16x16x4_f32(false, a, false, bfr,
                                                         (short)0, acc1, false, false);
            // source 1 target
            a.x = base.x * (m1.x * m1.x);
            a.y = base.y * (m1.y * m1.y);
            acc2 = __builtin_amdgcn_wmma_f32_16x16x4_f32(false, a, false, bfr,
                                                         (short)0, acc2, false, false);
            // source 1 noise
            {
                float u0 = 1.f - m1.x, u1 = 1.f - m1.y;
                a.x = base.x * (u0 * u0);
                a.y = base.y * (u1 * u1);
            }
            acc3 = __builtin_amdgcn_wmma_f32_16x16x4_f32(false, a, false, bfr,
                                                         (short)0, acc3, false, false);
        }
        __syncthreads();
    }

    // ---- spill the four 16x16 accumulators to LDS and fold into complex R ----
    float* tile = sm;   // 4*256 floats, reuse staging space
    const int mbase = (lane >> 4) << 3;   // rows 0..7 or 8..15
    const int ncol  = lane & 15;
    auto dump = [&](const v8f& A_, int mt) {
#pragma unroll
        for (int v = 0; v < 8; ++v)
            tile[mt * 256 + (mbase + v) * 16 + ncol] = A_[v];
    };
    dump(acc0, 0); dump(acc1, 1); dump(acc2, 2); dump(acc3, 3);
    __syncthreads();

    const float inv = 1.0f / (float)Tn;
    for (int idx = lane; idx < 256; idx += 32) {
        const int mt = idx >> 6, ij = idx & 63;
        const int i = ij >> 3, j = ij & 7;
        const int s = mt >> 1, which = mt & 1;       // 0 = target, 1 = noise
        const float re = (tile[mt * 256 + i * 16 + j] +
                          tile[mt * 256 + (i + 8) * 16 + (j + 8)]) * inv;
        const float im = (tile[mt * 256 + (i + 8) * 16 + j] -
                          tile[mt * 256 + i * 16 + (j + 8)]) * inv;
        const size_t o = (((size_t)(b * Sn + s) * Fn + f) * 2 + which) * 128 +
                         (size_t)(i * 8 + j) * 2;
        Rws[o]     = re;
        Rws[o + 1] = im;
    }
}

// ===================== Phase 2: per-problem dense solve ====================
struct cx { float r, i; };
__device__ __forceinline__ cx operator+(cx a, cx b) { return {a.r + b.r, a.i + b.i}; }
__device__ __forceinline__ cx operator-(cx a, cx b) { return {a.r - b.r, a.i - b.i}; }
__device__ __forceinline__ cx operator*(cx a, cx b) {
    return {a.r * b.r - a.i * b.i, a.r * b.i + a.i * b.r};
}
__device__ __forceinline__ cx cj(cx a)            { return {a.r, -a.i}; }
__device__ __forceinline__ cx smul(float s, cx a) { return {s * a.r, s * a.i}; }

__global__ __launch_bounds__(128)
void solve_kernel(const float* __restrict__ Rws, float* __restrict__ Wws)
{
    const int p = blockIdx.x * blockDim.x + threadIdx.x;
    if (p >= Bn * Sn * Fn) return;

    cx Rn_[8][8], Tm[8][8], Lm[8][8], A[8][8], U[8][8];
    const float* rt = Rws + (size_t)p * 256;
    const float* rn = rt + 128;
    for (int i = 0; i < 8; ++i)
        for (int j = 0; j < 8; ++j) {
            Tm[i][j]  = {rt[(i * 8 + j) * 2], rt[(i * 8 + j) * 2 + 1]};
            Rn_[i][j] = {rn[(i * 8 + j) * 2], rn[(i * 8 + j) * 2 + 1]};
        }

    // Cholesky R_n = L L^H (lower, real diagonal)
    for (int j = 0; j < 8; ++j) {
        float d = Rn_[j][j].r;
        for (int k = 0; k < j; ++k)
            d -= Lm[j][k].r * Lm[j][k].r + Lm[j][k].i * Lm[j][k].i;
        d = sqrtf(fmaxf(d, 1e-20f));
        Lm[j][j] = {d, 0.f};
        const float invd = 1.f / d;
        for (int i = j + 1; i < 8; ++i) {
            cx s_ = Rn_[i][j];
            for (int k = 0; k < j; ++k) s_ = s_ - Lm[i][k] * cj(Lm[j][k]);
            Lm[i][j] = smul(invd, s_);
        }
        for (int i = 0; i < j; ++i) Lm[i][j] = {0.f, 0.f};
    }

    // Tm <- L^{-1} R_t  (forward substitution, in place)
    for (int jc = 0; jc < 8; ++jc)
        for (int i = 0; i < 8; ++i) {
            cx s_ = Tm[i][jc];
            for (int k = 0; k < i; ++k) s_ = s_ - Lm[i][k] * Tm[k][jc];
            Tm[i][jc] = smul(1.f / Lm[i][i].r, s_);
        }

    // A <- L^{-1} Tm^H, then Hermitize: A = 0.5 (A + A^H)
    for (int jc = 0; jc < 8; ++jc)
        for (int i = 0; i < 8; ++i) {
            cx s_ = cj(Tm[jc][i]);
            for (int k = 0; k < i; ++k) s_ = s_ - Lm[i][k] * A[k][jc];
            A[i][jc] = smul(1.f / Lm[i][i].r, s_);
        }
    for (int i = 0; i < 8; ++i) {
        A[i][i].i = 0.f;
        for (int j = i + 1; j < 8; ++j) {
            cx h = smul(0.5f, A[i][j] + cj(A[j][i]));
            A[i][j] = h; A[j][i] = cj(h);
        }
    }

    // Hermitian Jacobi eigensolver, U accumulates eigenvectors (U unitary)
    for (int i = 0; i < 8; ++i)
        for (int j = 0; j < 8; ++j)
            U[i][j] = {(i == j) ? 1.f : 0.f, 0.f};

    for (int sweep = 0; sweep < 10; ++sweep)
        for (int pp = 0; pp < 7; ++pp)
            for (int q = pp + 1; q < 8; ++q) {
                const cx apq = A[pp][q];
                const float al = sqrtf(apq.r * apq.r + apq.i * apq.i);
                if (al < 1e-20f) continue;
                const cx ph = {apq.r / al, apq.i / al};
                const float tau = (A[q][q].r - A[pp][pp].r) / (2.f * al);
                const float t   = ((tau >= 0.f) ? 1.f : -1.f) /
                                  (fabsf(tau) + sqrtf(1.f + tau * tau));
                const float c   = 1.f / sqrtf(1.f + t * t);
                const float s   = t * c;
                const cx sph  = {s * ph.r,  s * ph.i};
                const cx scph = {s * ph.r, -s * ph.i};
                // A <- A * J
                for (int k = 0; k < 8; ++k) {
                    const cx akp = A[k][pp], akq = A[k][q];
                    A[k][pp] = smul(c, akp) - scph * akq;
                    A[k][q]  = sph * akp + smul(c, akq);
                }
                // A <- J^H * A
                for (int k = 0; k < 8; ++k) {
                    const cx apk = A[pp][k], aqk = A[q][k];
                    A[pp][k] = smul(c, apk) - sph * aqk;
                    A[q][k]  = scph * apk + smul(c, aqk);
                }
                // U <- U * J
                for (int k = 0; k < 8; ++k) {
                    const cx ukp = U[k][pp], ukq = U[k][q];
                    U[k][pp] = smul(c, ukp) - scph * ukq;
                    U[k][q]  = sph * ukp + smul(c, ukq);
                }
            }

    // top eigenvector (largest eigenvalue)
    int im = 0; float best = A[0][0].r;
    for (int i = 1; i < 8; ++i)
        if (A[i][i].r > best) { best = A[i][i].r; im = i; }
    cx u[8];
    for (int i = 0; i < 8; ++i) u[i] = U[i][im];

    // scale = (L * u_top)[ref_mic=0]   (U unitary => solve(V^H, e_last) = L U e_last)
    const cx scale = Lm[0][0] * u[0];

    // v_top = L^{-H} u_top  (back substitution on upper-triangular L^H)
    cx v[8];
    for (int i = 7; i >= 0; --i) {
        cx s_ = u[i];
        for (int k = i + 1; k < 8; ++k) s_ = s_ - cj(Lm[k][i]) * v[k];
        v[i] = smul(1.f / Lm[i][i].r, s_);
    }

    // w = conj(v_top) * scale
    float* wout = Wws + (size_t)p * 16;
    for (int cc = 0; cc < 8; ++cc) {
        const cx w = cj(v[cc]) * scale;
        wout[cc * 2]     = w.r;
        wout[cc * 2 + 1] = w.i;
    }
}

// ======================= Phase 3: apply beamformer =========================
// 250 threads/block, each handles 4 consecutive frames with b128 loads/stores.
__global__ __launch_bounds__(256)
void apply_kernel(const float* __restrict__ Xr,
                  const float* __restrict__ Xi,
                  const float* __restrict__ Wws,
                  float* __restrict__ out)
{
    const int bf = blockIdx.x;
    const int b  = bf / Fn;
    const int f  = bf % Fn;

    float w0r[8], w0i[8], w1r[8], w1i[8];
    const float* wp0 = Wws + ((size_t)(b * Sn + 0) * Fn + f) * 16;
    const float* wp1 = Wws + ((size_t)(b * Sn + 1) * Fn + f) * 16;
#pragma unroll
    for (int c = 0; c < 8; ++c) {
        w0r[c] = wp0[c * 2]; w0i[c] = wp0[c * 2 + 1];
        w1r[c] = wp1[c * 2]; w1i[c] = wp1[c * 2 + 1];
    }

    const int tid = threadIdx.x;
    if (tid >= Tn / 4) return;          // 250 active threads
    const int t0 = tid * 4;

    v4f y0re = {}, y0im = {}, y1re = {}, y1im = {};
#pragma unroll
    for (int c = 0; c < 8; ++c) {
        const size_t xo = ((size_t)(b * Cn + c) * Fn + f) * Tn + t0;
        const v4f xr = *(const v4f*)(Xr + xo);
        const v4f xi = *(const v4f*)(Xi + xo);
        y0re += w0r[c] * xr - w0i[c] * xi;
        y0im += w0r[c] * xi + w0i[c] * xr;
        y1re += w1r[c] * xr - w1i[c] * xi;
        y1im += w1r[c] * xi + w1i[c] * xr;
    }
    {
        float* o0 = out + 2 * (((size_t)(b * Sn + 0) * Fn + f) * Tn + t0);
        v4f lo, hi;
        lo.x = y0re.x; lo.y = y0im.x; lo.z = y0re.y; lo.w = y0im.y;
        hi.x = y0re.z; hi.y = y0im.z; hi.z = y0re.w; hi.w = y0im.w;
        *(v4f*)(o0)     = lo;
        *(v4f*)(o0 + 4) = hi;
    }
    {
        float* o1 = out + 2 * (((size_t)(b * Sn + 1) * Fn + f) * Tn + t0);
        v4f lo, hi;
        lo.x = y1re.x; lo.y = y1im.x; lo.z = y1re.y; lo.w = y1im.y;
        hi.x = y1re.z; hi.y = y1im.z; hi.z = y1re.w; hi.w = y1im.w;
        *(v4f*)(o1)     = lo;
        *(v4f*)(o1 + 4) = hi;
    }
}

// ================================ launcher =================================
extern "C" void kernel_launch(void* const* d_in, const int* in_sizes, int n_in,
                              void* d_out, int out_size, void* d_ws, size_t ws_size,
                              hipStream_t stream)
{
    const float* Xr = (const float*)d_in[0];
    const float* Xi = (const float*)d_in[1];
    const float* Mk = (const float*)d_in[2];
    float* out = (float*)d_out;

    float* Rws = (float*)d_ws;                              // (B*S*F) * 2 mats * 64 cx
    float* Wws = Rws + (size_t)Bn * Sn * Fn * 256;          // (B*S*F) * 8 cx weights

    cov_wmma_kernel<<<Bn * Fn, 32, 0, stream>>>(Xr, Xi, Mk, Rws);

    const int np = Bn * Sn * Fn;
    solve_kernel<<<(np + 127) / 128, 128, 0, stream>>>(Rws, Wws);

    apply_kernel<<<Bn * Fn, 256, 0, stream>>>(Xr, Xi, Wws, out);
}